// RNN_85607288144484
// MI455X (gfx1250) — compile-verified
//
#include <hip/hip_runtime.h>

typedef float v2f __attribute__((ext_vector_type(2)));
typedef float v4f __attribute__((ext_vector_type(4)));
typedef float v8f __attribute__((ext_vector_type(8)));
typedef int   v4i __attribute__((ext_vector_type(4)));

#define EPN  512
#define T1N  1024
#define NIN  128
#define HN   256
#define OUTD 64

// LDS row pitch (dwords): 16B-aligned for B128 async stores, and 36 mod 64
// banks gives conflict-free access for 16-lane fragment reads.
#define LPITCH 36

// ---------------------------------------------------------------------------
// fp32 WMMA: D(16x16) = A(16x4) * B(4x16) + C   -- V_WMMA_F32_16X16X4_F32
// A frag: lane<16 row=lane holds {K0,K1}; lane>=16 row=lane-16 holds {K2,K3}
// B frag: v0 = row K(kb), v1 = row K(kb+1), col = lane&15
// C/D:    v[i] -> (M = i + 8*(lane>=16), N = lane&15)
// ---------------------------------------------------------------------------
static __device__ __forceinline__ v8f wmma4(v2f a, v2f b, v8f c) {
  return __builtin_amdgcn_wmma_f32_16x16x4_f32(false, a, false, b, (short)0, c,
                                               false, false);
}

static __device__ __forceinline__ v8f v8f_zero() {
  v8f z = {0.f, 0.f, 0.f, 0.f, 0.f, 0.f, 0.f, 0.f};
  return z;
}

static __device__ __forceinline__ float sigmoidf_(float x) {
  return 1.0f / (1.0f + __expf(-x));
}
static __device__ __forceinline__ float tanhf_(float x) {
  float e = __expf(-2.0f * x);
  return (1.0f - e) / (1.0f + e);
}

// ---------------------------------------------------------------------------
// Global -> LDS staging via the CDNA5 async path (ASYNCcnt, bypasses VGPRs).
// Falls back to a v4f register copy if the builtin is unavailable.
// ---------------------------------------------------------------------------
#if __has_builtin(__builtin_amdgcn_global_load_async_to_lds_b128)
#define HAS_ASYNC_LDS 1
#else
#define HAS_ASYNC_LDS 0
#endif

static __device__ __forceinline__ void stage_b128(const float* gsrc,
                                                  float* ldst) {
#if HAS_ASYNC_LDS
  __builtin_amdgcn_global_load_async_to_lds_b128(
      (v4i __attribute__((address_space(1)))*)gsrc,
      (v4i __attribute__((address_space(3)))*)ldst, 0, 0);
#else
  *(v4f*)ldst = *(const v4f*)gsrc;
#endif
}

static __device__ __forceinline__ void wait_async_lds() {
#if HAS_ASYNC_LDS
#if __has_builtin(__builtin_amdgcn_s_wait_asynccnt)
  __builtin_amdgcn_s_wait_asynccnt(0);
#else
  asm volatile("s_wait_asynccnt 0" ::: "memory");
#endif
#endif
}

// ---------------------------------------------------------------------------
// P[o,h] = sum_t Wreg[o,t] * Hst[t,h]  for a 16-wide h chunk (rblk selects it)
// 8 waves split K=1024; LDS reduction. 64x16 output -> P (row-major 64x256).
// ---------------------------------------------------------------------------
static __device__ void reg_block(const float* __restrict__ Hst,
                                 const float* __restrict__ Wreg,
                                 float* __restrict__ P, int rblk, int tid,
                                 float* lds) {
  const int wave = tid >> 5, lane = tid & 31;
  const int l15 = lane & 15, kb = (lane >> 4) << 1;
  const int hcb = rblk << 4;  // 16 h-columns per block
  v8f acc[4];
#pragma unroll
  for (int rt = 0; rt < 4; ++rt) acc[rt] = v8f_zero();

  const int tb = wave << 7;  // 128 K per wave
  for (int ks = 0; ks < 32; ++ks) {
    const int k = tb + (ks << 2) + kb;
    v2f b = {Hst[(size_t)k * HN + hcb + l15],
             Hst[(size_t)(k + 1) * HN + hcb + l15]};
#pragma unroll
    for (int rt = 0; rt < 4; ++rt) {
      const float* ar = Wreg + (size_t)((rt << 4) + l15) * T1N;
      v2f a = {ar[k], ar[k + 1]};
      acc[rt] = wmma4(a, b, acc[rt]);
    }
  }
  const int rofs = (lane >> 4) * 8;
#pragma unroll
  for (int rt = 0; rt < 4; ++rt)
#pragma unroll
    for (int i = 0; i < 8; ++i)
      lds[(wave << 10) + (((rt << 4) + rofs + i) << 4) + l15] = acc[rt][i];
  __syncthreads();
  for (int j = tid; j < 1024; j += 256) {
    float s = 0.f;
#pragma unroll
    for (int w = 0; w < 8; ++w) s += lds[(w << 10) + j];
    P[(size_t)(j >> 4) * HN + hcb + (j & 15)] = s;
  }
}

// ---------------------------------------------------------------------------
// One GRU step. Blocks 0..63: 128x32 tile of h_new. Blocks 64..79: P[ep-1].
// ---------------------------------------------------------------------------
__global__ void __launch_bounds__(256) gru_step(
    const float* __restrict__ X, const float* __restrict__ Hin,
    float* __restrict__ Hout, const float* __restrict__ Wih,
    const float* __restrict__ Whh, const float* __restrict__ bih,
    const float* __restrict__ bhh, const float* __restrict__ Wreg,
    float* __restrict__ Pprev, int do_reg) {
  __shared__ float smem[8192];
  const int bid = blockIdx.x;
  const int tid = threadIdx.x;

  if (bid >= 64) {
    if (do_reg) reg_block(Hin, Wreg, Pprev, bid - 64, tid, smem);
    return;
  }

  const int wave = tid >> 5, lane = tid & 31;
  const int l15 = lane & 15, kb = (lane >> 4) << 1;
  const int bRow = (bid >> 3) << 7;  // 8 row blocks of 128 t-rows
  const int bCol = (bid & 7) << 5;   // 8 col blocks of 32 h-cols

  float* Xs = smem;                 // 128 x LPITCH
  float* Ws = smem + 128 * LPITCH;  // 3 gates x 32 x LPITCH

  // acc[0]=r(i+h), acc[1]=z(i+h), acc[2]=i_n, acc[3]=h_n ; 2 col tiles each
  v8f acc[4][2];
#pragma unroll
  for (int g = 0; g < 4; ++g)
#pragma unroll
    for (int ct = 0; ct < 2; ++ct) acc[g][ct] = v8f_zero();

#pragma unroll
  for (int phase = 0; phase < 2; ++phase) {
    const float* src = phase ? Hin : X;
    const float* W = phase ? Whh : Wih;
    const int ld = phase ? HN : NIN;  // also the K extent
    for (int kc = 0; kc < ld; kc += 32) {
      __syncthreads();
      // stage src tile: 128 rows x 32 k = 1024 x b128
      for (int idx = tid; idx < 1024; idx += 256) {
        const int r0 = idx >> 3, q = (idx & 7) << 2;
        stage_b128(&src[(size_t)(bRow + r0) * ld + kc + q],
                   &Xs[r0 * LPITCH + q]);
      }
      // stage weights: 3 gates x 32 n x 32 k = 768 x b128
      for (int idx = tid; idx < 768; idx += 256) {
        const int g = idx >> 8, rem = idx & 255, n = rem >> 3,
                  q = (rem & 7) << 2;
        stage_b128(&W[(size_t)(g * HN + bCol + n) * ld + kc + q],
                   &Ws[(g * 32 + n) * LPITCH + q]);
      }
      wait_async_lds();
      __syncthreads();
      const float* xr = &Xs[(wave * 16 + l15) * LPITCH];
#pragma unroll
      for (int kk = 0; kk < 8; ++kk) {
        const int k0 = (kk << 2) + kb;
        v2f a = {xr[k0], xr[k0 + 1]};
#pragma unroll
        for (int g = 0; g < 3; ++g) {
          const int gi = (phase && g == 2) ? 3 : g;
#pragma unroll
          for (int ct = 0; ct < 2; ++ct) {
            const float* wr = &Ws[(g * 32 + ct * 16 + l15) * LPITCH];
            v2f b = {wr[k0], wr[k0 + 1]};
            acc[gi][ct] = wmma4(a, b, acc[gi][ct]);
          }
        }
      }
    }
  }

  // gate math + h update
  const int rofs = (lane >> 4) * 8;
#pragma unroll
  for (int ct = 0; ct < 2; ++ct) {
    const int c = bCol + ct * 16 + l15;
    const float br = bih[c] + bhh[c];
    const float bz = bih[256 + c] + bhh[256 + c];
    const float bin = bih[512 + c];
    const float bhn = bhh[512 + c];
#pragma unroll
    for (int i = 0; i < 8; ++i) {
      const int t = bRow + wave * 16 + rofs + i;
      const float hp = Hin[(size_t)t * HN + c];
      const float r = sigmoidf_(acc[0][ct][i] + br);
      const float z = sigmoidf_(acc[1][ct][i] + bz);
      const float nt = tanhf_(acc[2][ct][i] + bin + r * (acc[3][ct][i] + bhn));
      Hout[(size_t)t * HN + c] = (1.f - z) * nt + z * hp;
    }
  }
}

__global__ void __launch_bounds__(256) gru_reg_tail(
    const float* __restrict__ Hst, const float* __restrict__ Wreg,
    float* __restrict__ P) {
  __shared__ float lds[8192];
  reg_block(Hst, Wreg, P, blockIdx.x, threadIdx.x, lds);
}

// ---------------------------------------------------------------------------
// final[ep,n,o] = sum_h Whr[n,h]*P[ep,o,h] + bhr[n]*SW[o] + breg[o]
// ---------------------------------------------------------------------------
__global__ void __launch_bounds__(256) gru_final(
    const float* __restrict__ P, const float* __restrict__ Whr,
    const float* __restrict__ bhr, const float* __restrict__ breg,
    const float* __restrict__ SW, float* __restrict__ out) {
  const int ep = blockIdx.x;
  const int tid = threadIdx.x;
  const int wave = tid >> 5, lane = tid & 31;
  const int l15 = lane & 15, kb = (lane >> 4) << 1;
  const float* Pe = P + (size_t)ep * OUTD * HN;
  v8f acc[4];
#pragma unroll
  for (int ct = 0; ct < 4; ++ct) acc[ct] = v8f_zero();

  const float* ar = Whr + (size_t)(wave * 16 + l15) * HN;
  for (int h4 = 0; h4 < HN / 4; ++h4) {
    const int h = (h4 << 2) + kb;
    v2f a = {ar[h], ar[h + 1]};
#pragma unroll
    for (int ct = 0; ct < 4; ++ct) {
      const float* br = Pe + (size_t)((ct << 4) + l15) * HN;
      v2f b = {br[h], br[h + 1]};
      acc[ct] = wmma4(a, b, acc[ct]);
    }
  }
  const int rofs = (lane >> 4) * 8;
#pragma unroll
  for (int ct = 0; ct < 4; ++ct) {
    const int o = (ct << 4) + l15;
    const float bo = breg[o], sw = SW[o];
#pragma unroll
    for (int i = 0; i < 8; ++i) {
      const int n = wave * 16 + rofs + i;
      out[(size_t)ep * NIN * OUTD + (size_t)n * OUTD + o] =
          acc[ct][i] + bhr[n] * sw + bo;
    }
  }
}

__global__ void __launch_bounds__(256) gru_init(const float* __restrict__ hidden,
                                                float* __restrict__ h0,
                                                const float* __restrict__ Wreg,
                                                float* __restrict__ SW) {
  if (blockIdx.x < 128) {
    const int base = blockIdx.x * 2048;
    for (int i = threadIdx.x; i < 2048; i += 256) h0[base + i] = hidden[base + i];
  } else {
    const int o = threadIdx.x;
    if (o < OUTD) {
      float s = 0.f;
      for (int t = 0; t < T1N; ++t) s += Wreg[(size_t)o * T1N + t];
      SW[o] = s;
    }
  }
}

// ---------------------------------------------------------------------------
extern "C" void kernel_launch(void* const* d_in, const int* in_sizes, int n_in,
                              void* d_out, int out_size, void* d_ws,
                              size_t ws_size, hipStream_t stream) {
  const float* input  = (const float*)d_in[0];
  const float* hidden = (const float*)d_in[1];
  const float* W_ih   = (const float*)d_in[2];
  const float* W_hh   = (const float*)d_in[3];
  const float* b_ih   = (const float*)d_in[4];
  const float* b_hh   = (const float*)d_in[5];
  const float* W_hr   = (const float*)d_in[6];
  const float* b_hr   = (const float*)d_in[7];
  const float* W_reg  = (const float*)d_in[8];
  const float* b_reg  = (const float*)d_in[9];
  float* out = (float*)d_out;

  // workspace: hA | hB | P[512][64][256] | SW[64]  (~35.7 MB)
  float* ws = (float*)d_ws;
  float* hA = ws;
  float* hB = ws + (size_t)T1N * HN;
  float* P  = ws + (size_t)2 * T1N * HN;
  float* SW = P + (size_t)EPN * OUTD * HN;

  gru_init<<<129, 256, 0, stream>>>(hidden, hA, W_reg, SW);

  for (int ep = 0; ep < EPN; ++ep) {
    const float* hp = (ep & 1) ? hB : hA;
    float* hn       = (ep & 1) ? hA : hB;
    const float* Xe = input + (size_t)ep * T1N * NIN;
    float* Pprev = (ep > 0) ? (P + (size_t)(ep - 1) * OUTD * HN) : P;
    gru_step<<<80, 256, 0, stream>>>(Xe, hp, hn, W_ih, W_hh, b_ih, b_hh, W_reg,
                                     Pprev, ep > 0 ? 1 : 0);
  }
  // Hout(511) lives in hA (511 is odd -> hn == hA)
  float* hlast = ((EPN - 1) & 1) ? hA : hB;
  gru_reg_tail<<<16, 256, 0, stream>>>(hlast, W_reg,
                                       P + (size_t)(EPN - 1) * OUTD * HN);
  gru_final<<<EPN, 256, 0, stream>>>(P, W_hr, b_hr, b_reg, SW, out);
}